// CRF_19172734009874
// MI455X (gfx1250) — compile-verified
//
#include <hip/hip_runtime.h>
#include <hip/hip_bf16.h>

typedef float v2f __attribute__((ext_vector_type(2)));
typedef float v8f __attribute__((ext_vector_type(8)));

#define BATCH   1024
#define S_LEN   512
#define NTAG    48
#define TDIM    50          // transitions are (NTAG+2) x (NTAG+2)
#define START_T 48
#define END_T   49

// ---------------------------------------------------------------------------
// Kernel 1: numerator (joint log-likelihood), one wave32 per batch.
// ---------------------------------------------------------------------------
__global__ __launch_bounds__(32) void crf_num_kernel(
    const float* __restrict__ emis, const int* __restrict__ tags,
    const float* __restrict__ trans, float* __restrict__ num)
{
    const int b    = blockIdx.x;
    const int lane = threadIdx.x;

    float part = 0.0f;
    float msum = 0.0f;
    for (int s = lane; s < S_LEN; s += 32) {
        int   tg = tags[b * S_LEN + s];
        int   st = tg < 0 ? 0 : tg;
        float m  = (tg != -1) ? 1.0f : 0.0f;
        float em = emis[(b * S_LEN + s) * NTAG + st];
        if (s == 0) {
            part += trans[START_T * TDIM + st] + em * m;
        } else {
            int tp = tags[b * S_LEN + s - 1];
            tp = tp < 0 ? 0 : tp;
            part += (em + trans[tp * TDIM + st]) * m;
        }
        msum += m;
    }
    for (int off = 1; off < 32; off <<= 1) {
        part += __shfl_xor(part, off, 32);
        msum += __shfl_xor(msum, off, 32);
    }
    if (lane == 0) {
        int last = (int)msum - 1;
        if (last < 0) last = 0;
        int lt = tags[b * S_LEN + last];
        lt = lt < 0 ? 0 : lt;
        num[b] = part + trans[lt * TDIM + END_T];
    }
}

// ---------------------------------------------------------------------------
// Kernel 2: log-partition forward scan via f32 WMMA.
// One wave32 per 16-batch tile; lp kept in WMMA C-layout across 3 N-tiles.
// Step: lp' = em + rmax + log( exp(lp - rmax) x expT )   (GEMM 16x48x48)
// Split-K double accumulators halve the dependent-WMMA chain (6 deep x 6).
// ---------------------------------------------------------------------------
__global__ __launch_bounds__(32) void crf_scan_kernel(
    const float* __restrict__ emis, const int* __restrict__ tags,
    const float* __restrict__ trans, float* __restrict__ den)
{
    __shared__ float xlds[16 * 49];          // padded stride: no bank conflicts

    const int lane = threadIdx.x;
    const int half = lane >> 4;              // 0 or 1
    const int ln   = lane & 15;
    const int b0   = blockIdx.x * 16;

    // --- B fragments: expT in 16x16x4 B-layout, resident in VGPRs ----------
    // element (k, n*16+ln): vgpr0 holds k = kc*4 + 2*half, vgpr1 = +1
    v2f Bf[12][3];
#pragma unroll
    for (int kc = 0; kc < 12; ++kc) {
        const int krow = kc * 4 + 2 * half;
#pragma unroll
        for (int n = 0; n < 3; ++n) {
            const int j = n * 16 + ln;
            Bf[kc][n].x = __expf(trans[(krow + 0) * TDIM + j]);
            Bf[kc][n].y = __expf(trans[(krow + 1) * TDIM + j]);
        }
    }

    // --- init lp in C/D layout: lane holds (m = r + 8*half, j = n*16+ln) ---
    v8f lp[3];
#pragma unroll
    for (int n = 0; n < 3; ++n) {
        const int j = n * 16 + ln;
#pragma unroll
        for (int r = 0; r < 8; ++r) {
            const int b = b0 + r + 8 * half;
            lp[n][r] = emis[(b * S_LEN) * NTAG + j] + trans[START_T * TDIM + j];
        }
    }

    for (int t = 1; t < S_LEN; ++t) {
        // prefetch next step's emissions rows into cache
        if (t + 1 < S_LEN) {
            const char* p =
                (const char*)&emis[((b0 + ln) * S_LEN + (t + 1)) * NTAG];
            __builtin_prefetch(p + half * 128, 0, 0);
        }

        // hoist this step's tag loads so latency hides under the matmul
        int tg[8];
#pragma unroll
        for (int r = 0; r < 8; ++r)
            tg[r] = tags[(b0 + r + 8 * half) * S_LEN + t];

        // per-batch row max over 48 tags (3 tiles x 16 lanes of this half)
        float rmax[8];
#pragma unroll
        for (int r = 0; r < 8; ++r) {
            float v = fmaxf(fmaxf(lp[0][r], lp[1][r]), lp[2][r]);
            v = fmaxf(v, __shfl_xor(v, 1, 32));
            v = fmaxf(v, __shfl_xor(v, 2, 32));
            v = fmaxf(v, __shfl_xor(v, 4, 32));
            v = fmaxf(v, __shfl_xor(v, 8, 32));
            rmax[r] = v;
        }

        __syncthreads();                     // WAR vs. previous step's reads
#pragma unroll
        for (int n = 0; n < 3; ++n)
#pragma unroll
            for (int r = 0; r < 8; ++r)
                xlds[(r + 8 * half) * 49 + (n * 16 + ln)] =
                    __expf(lp[n][r] - rmax[r]);
        __syncthreads();

        // GEMM: acc = exp(lp - rmax) x expT, 36 x v_wmma_f32_16x16x4_f32,
        // split-K into two 6-deep independent accumulation chains per tile.
        v8f acc0[3] = {};
        v8f acc1[3] = {};
#pragma unroll
        for (int kc = 0; kc < 6; ++kc) {
            const int kbA = kc * 4 + 2 * half;
            const int kbB = (kc + 6) * 4 + 2 * half;
            v2f Aa, Ab;
            Aa.x = xlds[ln * 49 + kbA + 0];
            Aa.y = xlds[ln * 49 + kbA + 1];
            Ab.x = xlds[ln * 49 + kbB + 0];
            Ab.y = xlds[ln * 49 + kbB + 1];
#pragma unroll
            for (int n = 0; n < 3; ++n) {
                acc0[n] = __builtin_amdgcn_wmma_f32_16x16x4_f32(
                    false, Aa, false, Bf[kc][n], (short)0, acc0[n],
                    false, false);
                acc1[n] = __builtin_amdgcn_wmma_f32_16x16x4_f32(
                    false, Ab, false, Bf[kc + 6][n], (short)0, acc1[n],
                    false, false);
            }
        }
        v8f acc[3];
#pragma unroll
        for (int n = 0; n < 3; ++n) acc[n] = acc0[n] + acc1[n];

        // lp' = em + rmax + log(acc), carried through where mask == 0
#pragma unroll
        for (int r = 0; r < 8; ++r) {
            const int b = b0 + r + 8 * half;
#pragma unroll
            for (int n = 0; n < 3; ++n) {
                const int j    = n * 16 + ln;
                const float em = emis[(b * S_LEN + t) * NTAG + j];
                const float sc = em + rmax[r] + __logf(acc[n][r]);
                lp[n][r] = (tg[r] != -1) ? sc : lp[n][r];
            }
        }
    }

    // finalize: den[b] = logsumexp_j( lp[b,j] + T[j, END] )
#pragma unroll
    for (int n = 0; n < 3; ++n) {
        const int j = n * 16 + ln;
#pragma unroll
        for (int r = 0; r < 8; ++r)
            lp[n][r] += trans[j * TDIM + END_T];
    }
#pragma unroll
    for (int r = 0; r < 8; ++r) {
        float v = fmaxf(fmaxf(lp[0][r], lp[1][r]), lp[2][r]);
        v = fmaxf(v, __shfl_xor(v, 1, 32));
        v = fmaxf(v, __shfl_xor(v, 2, 32));
        v = fmaxf(v, __shfl_xor(v, 4, 32));
        v = fmaxf(v, __shfl_xor(v, 8, 32));
        float s = __expf(lp[0][r] - v) + __expf(lp[1][r] - v) +
                  __expf(lp[2][r] - v);
        s += __shfl_xor(s, 1, 32);
        s += __shfl_xor(s, 2, 32);
        s += __shfl_xor(s, 4, 32);
        s += __shfl_xor(s, 8, 32);
        if (ln == 0)
            den[b0 + r + 8 * half] = v + __logf(s);
    }
}

// ---------------------------------------------------------------------------
// Kernel 3: deterministic tree reduction -> mean(num - den)
// ---------------------------------------------------------------------------
__global__ __launch_bounds__(256) void crf_reduce_kernel(
    const float* __restrict__ num, const float* __restrict__ den,
    float* __restrict__ out)
{
    __shared__ float sm[256];
    const int tid = threadIdx.x;
    float s = 0.0f;
    for (int i = tid; i < BATCH; i += 256) s += num[i] - den[i];
    sm[tid] = s;
    __syncthreads();
    for (int w = 128; w > 0; w >>= 1) {
        if (tid < w) sm[tid] += sm[tid + w];
        __syncthreads();
    }
    if (tid == 0) out[0] = sm[0] * (1.0f / (float)BATCH);
}

// ---------------------------------------------------------------------------
extern "C" void kernel_launch(void* const* d_in, const int* in_sizes, int n_in,
                              void* d_out, int out_size, void* d_ws,
                              size_t ws_size, hipStream_t stream)
{
    const float* emis  = (const float*)d_in[0];   // (1024, 512, 48) f32
    const int*   tags  = (const int*)d_in[1];     // (1024, 512) i32
    const float* trans = (const float*)d_in[2];   // (50, 50) f32
    float* out = (float*)d_out;                   // scalar

    float* num = (float*)d_ws;                    // [1024]
    float* den = num + BATCH;                     // [1024]

    crf_num_kernel<<<BATCH, 32, 0, stream>>>(emis, tags, trans, num);
    crf_scan_kernel<<<BATCH / 16, 32, 0, stream>>>(emis, tags, trans, den);
    crf_reduce_kernel<<<1, 256, 0, stream>>>(num, den, out);
}